// SineSynth_30958124269904
// MI455X (gfx1250) — compile-verified
//
#include <hip/hip_runtime.h>
#include <math.h>

// ---------------- problem constants ----------------
#define B_      4
#define S_      1000
#define F_      256
#define R_      32
#define T_      (F_ * R_)           // 8192 samples
#define FPB     2                   // frames per block (kernel 2)
#define CHUNKS  (F_ / FPB)          // 128 chunks per batch

#define INV_FS      2.2675736961451248e-5f   // 1 / 44100  (revolutions per Hz-sample)
#define INV_TWO_PI  0.15915494309189533577f

// ---------------- CDNA5 async global->LDS helpers ----------------
// Low 32 bits of a generic LDS address are the LDS byte offset
// (aperture bits live in the high dword on CDNA5).
__device__ __forceinline__ void async_ld_b128(void* lds_dst, const void* gsrc) {
    unsigned ldsa = (unsigned)(size_t)lds_dst;
    unsigned long long ga = (unsigned long long)(size_t)gsrc;
    asm volatile("global_load_async_to_lds_b128 %0, %1, off"
                 :: "v"(ldsa), "v"(ga)
                 : "memory");
}

__device__ __forceinline__ void wait_async_all() {
#if __has_builtin(__builtin_amdgcn_s_wait_asynccnt)
    __builtin_amdgcn_s_wait_asynccnt(0);
#else
    asm volatile("s_wait_asynccnt 0" ::: "memory");
#endif
}

// ---------------- kernel 1: per-(b,s) frame prefix + transpose ----------------
// Frame-sum closed form of the 32x linear upsample (align_corners=False):
//   W_k = 4*f[k-1] + 24*f[k] + 4*f[k+1]   (indices clamped at the edges,
//   reproducing the reference's coordinate clipping exactly).
// Pb[b,k,s] = (init_phase/2pi) + (1/FS) * sum_{k'<k} W_k'  in REVOLUTIONS,
// wrapped mod 1 (sin-invariant, keeps fp32 accuracy and v_sin range tiny).
// Also transposes f,a into [b,k,s] so kernel 2's sine loop is contiguous.
__global__ void sine_prep_kernel(const float* __restrict__ freq,
                                 const float* __restrict__ amp,
                                 const float* __restrict__ phase0,
                                 float* __restrict__ Ft,
                                 float* __restrict__ At,
                                 float* __restrict__ Pb) {
    int idx = blockIdx.x * blockDim.x + threadIdx.x;
    if (idx >= B_ * S_) return;
    int b = idx / S_;
    int s = idx - b * S_;
    const float* f = freq + (size_t)idx * F_;
    const float* a = amp  + (size_t)idx * F_;

    float run = phase0[idx] * INV_TWO_PI;        // radians -> revolutions
    for (int k = 0; k < F_; ++k) {
        float fk  = f[k];
        float fm1 = f[(k == 0)      ? 0       : k - 1];
        float fp1 = f[(k == F_ - 1) ? F_ - 1  : k + 1];
        size_t o = ((size_t)(b * F_ + k)) * S_ + s;
        Ft[o] = fk;
        At[o] = a[k];
        Pb[o] = run;
        float W = fmaf(24.0f, fk, 4.0f * (fm1 + fp1));
        run += W * INV_FS;
        run -= floorf(run);                      // wrap mod 1 revolution
    }
}

// ---------------- kernel 2: additive synthesis ----------------
// block = 2 waves (wave32); wave -> one frame k; lane -> sample t = 32k+lane.
// The in-frame inclusive phase scan is a fixed linear functional of the three
// wave-uniform frame values, so the cross-lane scan is done ONCE at setup on
// the coefficient vectors (15 shuffles), not per sine. Inner loop per 4 sines:
// six ds_load_b128 + ~28 VALU + 4 v_sin_f32, register accumulation only.
__global__ __launch_bounds__(FPB * 32)
void sine_synth_kernel(const float* __restrict__ Ft,
                       const float* __restrict__ At,
                       const float* __restrict__ Pb,
                       float* __restrict__ out) {
    __shared__ __align__(16) float smem[(FPB + 2) * S_ * 2 + FPB * S_]; // 40 KB
    float* lf = smem;                      // (FPB+2) frame slots of freq
    float* la = smem + (FPB + 2) * S_;     // (FPB+2) frame slots of amp
    float* lp = smem + 2 * (FPB + 2) * S_; // FPB slots of phase base (revs)

    const int tid  = threadIdx.x;
    const int lane = tid & 31;
    const int wv   = tid >> 5;
    const int b     = blockIdx.x / CHUNKS;
    const int chunk = blockIdx.x - b * CHUNKS;
    const int k0    = chunk * FPB;

    // ---- async stage: f/a frames [k0-1 .. k0+FPB] (clamped), p frames [k0 .. k0+FPB-1]
    const int Q = S_ / 4;  // b128 chunks per frame row (S_ % 4 == 0)
    for (int i = tid; i < (FPB + 2) * Q; i += blockDim.x) {
        int slot = i / Q;
        int c4   = (i - slot * Q) * 4;
        int gf   = k0 - 1 + slot;
        gf = gf < 0 ? 0 : (gf > F_ - 1 ? F_ - 1 : gf);
        size_t go = ((size_t)(b * F_ + gf)) * S_ + c4;
        async_ld_b128(&lf[slot * S_ + c4], &Ft[go]);
        async_ld_b128(&la[slot * S_ + c4], &At[go]);
    }
    for (int i = tid; i < FPB * Q; i += blockDim.x) {
        int slot = i / Q;
        int c4   = (i - slot * Q) * 4;
        size_t go = ((size_t)(b * F_ + k0 + slot)) * S_ + c4;
        async_ld_b128(&lp[slot * S_ + c4], &Pb[go]);
    }

    // ---- per-lane interpolation constants (match reference coords exactly)
    const int k = k0 + wv;
    float c = (float)k - 0.5f + ((float)lane + 0.5f) * (1.0f / 32.0f);
    c = fminf(fmaxf(c, 0.0f), (float)(F_ - 1));
    int   lo = (int)c;                        // c >= 0
    float w  = c - (float)lo;
    int   hi = (lo + 1 > F_ - 1) ? (F_ - 1) : lo + 1;
    const int slotLo = lo - (k0 - 1);         // LDS frame slots in {wv..wv+2}
    const int slotHi = hi - (k0 - 1);
    const int sA = wv, sB = wv + 1, sC = wv + 2;

    // per-lane contribution of this lane's omega to the three canonical rows
    float eA = ((slotLo == sA) ? (1.0f - w) : 0.0f) + ((slotHi == sA) ? w : 0.0f);
    float eB = ((slotLo == sB) ? (1.0f - w) : 0.0f) + ((slotHi == sB) ? w : 0.0f);
    float eC = ((slotLo == sC) ? (1.0f - w) : 0.0f) + ((slotHi == sC) ? w : 0.0f);
    // ONE inclusive wave32 scan of the coefficient vectors (replaces per-sine scans)
    #pragma unroll
    for (int off = 1; off < 32; off <<= 1) {
        float tA = __shfl_up(eA, (unsigned)off, 32);
        float tB = __shfl_up(eB, (unsigned)off, 32);
        float tC = __shfl_up(eC, (unsigned)off, 32);
        if (lane >= off) { eA += tA; eB += tB; eC += tC; }
    }
    const float alpha = eA * INV_FS;          // phase (revolutions) = pb +
    const float beta  = eB * INV_FS;          //   alpha*f[kA] + beta*f[kB] + gamma*f[kC]
    const float gamma = eC * INV_FS;

    wait_async_all();
    __syncthreads();

    // float4 row pointers (rows are 4000 B apart -> 16 B aligned)
    const float4* lfA  = (const float4*)(lf + sA * S_);
    const float4* lfB  = (const float4*)(lf + sB * S_);
    const float4* lfC  = (const float4*)(lf + sC * S_);
    const float4* laLo = (const float4*)(la + slotLo * S_);
    const float4* laHi = (const float4*)(la + slotHi * S_);
    const float4* lpp  = (const float4*)(lp + wv * S_);

    float acc = 0.0f;
    for (int q = 0; q < S_ / 4; ++q) {
        float4 fA = lfA[q], fB = lfB[q], fC = lfC[q];
        float4 aL = laLo[q], aH = laHi[q];
        float4 pb = lpp[q];

        float ph0 = fmaf(gamma, fC.x, fmaf(beta, fB.x, fmaf(alpha, fA.x, pb.x)));
        float ph1 = fmaf(gamma, fC.y, fmaf(beta, fB.y, fmaf(alpha, fA.y, pb.y)));
        float ph2 = fmaf(gamma, fC.z, fmaf(beta, fB.z, fmaf(alpha, fA.z, pb.z)));
        float ph3 = fmaf(gamma, fC.w, fmaf(beta, fB.w, fmaf(alpha, fA.w, pb.w)));

        float a0 = fmaf(w, aH.x - aL.x, aL.x);
        float a1 = fmaf(w, aH.y - aL.y, aL.y);
        float a2 = fmaf(w, aH.z - aL.z, aL.z);
        float a3 = fmaf(w, aH.w - aL.w, aL.w);

        acc = fmaf(a0, __builtin_amdgcn_sinf(ph0), acc);  // v_sin_f32: sin(2*pi*x)
        acc = fmaf(a1, __builtin_amdgcn_sinf(ph1), acc);
        acc = fmaf(a2, __builtin_amdgcn_sinf(ph2), acc);
        acc = fmaf(a3, __builtin_amdgcn_sinf(ph3), acc);
    }
    out[(size_t)b * T_ + (size_t)k * 32 + lane] = acc;
}

// ---------------- host launch ----------------
extern "C" void kernel_launch(void* const* d_in, const int* in_sizes, int n_in,
                              void* d_out, int out_size, void* d_ws, size_t ws_size,
                              hipStream_t stream) {
    const float* freq = (const float*)d_in[0];  // [B,S,F]
    const float* amp  = (const float*)d_in[1];  // [B,S,F]
    const float* ph0  = (const float*)d_in[2];  // [B,S]
    float* out = (float*)d_out;                 // [B,1,T]

    float* W  = (float*)d_ws;                   // needs 3*B*F*S floats (~12 MB)
    const size_t plane = (size_t)B_ * F_ * S_;
    float* Ft = W;
    float* At = W + plane;
    float* Pb = W + 2 * plane;

    sine_prep_kernel<<<(B_ * S_ + 255) / 256, 256, 0, stream>>>(
        freq, amp, ph0, Ft, At, Pb);
    sine_synth_kernel<<<B_ * CHUNKS, FPB * 32, 0, stream>>>(
        Ft, At, Pb, out);
}